// MultiBoxLoss_24953759990299
// MI455X (gfx1250) — compile-verified
//
#include <hip/hip_runtime.h>
#include <hip/hip_bf16.h>
#include <math.h>

#define BATCH 128
#define P 8732
#define M 16
#define NCLS 3
#define NT 256
#define THRESH 0.5f

typedef __attribute__((ext_vector_type(2))) float v2f;
typedef __attribute__((ext_vector_type(8))) float v8f;

__device__ __forceinline__ float bce_logit(float x, float t) {
    // logaddexp(0,x) - x*t  (numerically stable BCE-with-logits)
    return fmaxf(x, 0.0f) + log1pf(expf(-fabsf(x))) - x * t;
}

__device__ __forceinline__ float blk_reduce(float v, float* buf) {
    int tid = threadIdx.x;
    buf[tid] = v;
    __syncthreads();
    for (int s = NT / 2; s > 0; s >>= 1) {
        if (tid < s) buf[tid] += buf[tid + s];
        __syncthreads();
    }
    float r = buf[0];
    __syncthreads();
    return r;
}

// One workgroup per batch image: match priors<->objects, compute per-prior
// losses, and hard-negative top-k sum via bit-pattern binary search (no sort).
__global__ void __launch_bounds__(NT)
match_loss_kernel(const float* __restrict__ locs,    // (B,P,4)
                  const float* __restrict__ scores,  // (B,P,3)
                  const float* __restrict__ boxes,   // (B,M,4) xyxy
                  const int*   __restrict__ labels,  // (B,M)
                  const float* __restrict__ priors,  // (P,4) cxcywh
                  float* __restrict__ ws)            // [0..127]=conf_pos, [128..]=l1, [256..]=hardneg, [384..]=npos
{
    const int b = blockIdx.x;
    const int tid = threadIdx.x;

    __shared__ float s_box[M * 4];
    __shared__ float s_area[M];
    __shared__ int   s_lab[M];
    __shared__ int   s_prior_obj[M];
    __shared__ short s_objp[P];   // low 5 bits: object id, bit5: positive-overlap flag
    __shared__ float s_row[P];    // conf_neg values
    __shared__ float s_redv[NT];
    __shared__ int   s_redi[NT];

    if (tid < M * 4) s_box[tid] = boxes[(size_t)b * M * 4 + tid];
    if (tid < M)     s_lab[tid] = labels[b * M + tid];
    __syncthreads();
    if (tid < M) {
        float x0 = s_box[tid * 4 + 0], y0 = s_box[tid * 4 + 1];
        float x1 = s_box[tid * 4 + 2], y1 = s_box[tid * 4 + 3];
        s_area[tid] = (x1 - x0) * (y1 - y0);
    }
    __syncthreads();

    // ---- Phase 1: per-prior best object; per-thread per-object best prior ----
    float bestv[M];
    int   bestp[M];
#pragma unroll
    for (int m = 0; m < M; ++m) { bestv[m] = -1.0f; bestp[m] = P; }

    for (int p = tid; p < P; p += NT) {
        float pcx = priors[p * 4 + 0], pcy = priors[p * 4 + 1];
        float pw  = priors[p * 4 + 2], ph  = priors[p * 4 + 3];
        float px0 = pcx - 0.5f * pw, py0 = pcy - 0.5f * ph;
        float px1 = pcx + 0.5f * pw, py1 = pcy + 0.5f * ph;
        float pa = (px1 - px0) * (py1 - py0);
        float bov = -1.0f; int bm = 0;
#pragma unroll
        for (int m = 0; m < M; ++m) {
            float ix0 = fmaxf(s_box[m * 4 + 0], px0);
            float iy0 = fmaxf(s_box[m * 4 + 1], py0);
            float ix1 = fminf(s_box[m * 4 + 2], px1);
            float iy1 = fminf(s_box[m * 4 + 3], py1);
            float iw = fmaxf(ix1 - ix0, 0.0f), ih = fmaxf(iy1 - iy0, 0.0f);
            float inter = iw * ih;
            float iou = inter / (s_area[m] + pa - inter);
            if (iou > bov) { bov = iou; bm = m; }                 // first-max (m ascending)
            if (iou > bestv[m]) { bestv[m] = iou; bestp[m] = p; } // first-max (p ascending per thread)
        }
        s_objp[p] = (short)(bm | (bov >= THRESH ? 32 : 0));
    }
    __syncthreads();

    // ---- Phase 2: per-object argmax over priors (tie-break: smaller p) ----
    for (int m = 0; m < M; ++m) {
        s_redv[tid] = bestv[m];
        s_redi[tid] = bestp[m];
        __syncthreads();
        for (int s = NT / 2; s > 0; s >>= 1) {
            if (tid < s) {
                float v1 = s_redv[tid], v2 = s_redv[tid + s];
                int   p1 = s_redi[tid], p2 = s_redi[tid + s];
                if (v2 > v1 || (v2 == v1 && p2 < p1)) { s_redv[tid] = v2; s_redi[tid] = p2; }
            }
            __syncthreads();
        }
        if (tid == 0) s_prior_obj[m] = s_redi[0];
        __syncthreads();
    }
    if (tid == 0) {
        for (int m = 0; m < M; ++m)
            s_objp[s_prior_obj[m]] = (short)(m | 32);  // forced match, overlap := 1.0 (>= thresh)
    }
    __syncthreads();

    // ---- Phase 3: per-prior BCE + masked L1; fill conf_neg row ----
    float accC = 0.0f, accL = 0.0f, accN = 0.0f;
    for (int p = tid; p < P; p += NT) {
        int e = s_objp[p];
        int m = e & 31;
        int lab = (e & 32) ? s_lab[m] : 0;
        const float* sc = scores + ((size_t)b * P + p) * NCLS;
        float t0 = (lab == 0) ? 1.0f : 0.0f;
        float t1 = (lab == 1 || lab == 3) ? 1.0f : 0.0f;
        float t2 = (lab == 2 || lab == 3) ? 1.0f : 0.0f;
        float bce = bce_logit(sc[0], t0) + bce_logit(sc[1], t1) + bce_logit(sc[2], t2);
        if (lab > 0) {
            float bx0 = s_box[m * 4 + 0], by0 = s_box[m * 4 + 1];
            float bx1 = s_box[m * 4 + 2], by1 = s_box[m * 4 + 3];
            float bcx = 0.5f * (bx0 + bx1), bcy = 0.5f * (by0 + by1);
            float bw = bx1 - bx0, bh = by1 - by0;
            float pcx = priors[p * 4 + 0], pcy = priors[p * 4 + 1];
            float pw  = priors[p * 4 + 2], ph  = priors[p * 4 + 3];
            float g0 = (bcx - pcx) / (pw * 0.1f);
            float g1 = (bcy - pcy) / (ph * 0.1f);
            float g2 = logf(bw / pw) * 5.0f;
            float g3 = logf(bh / ph) * 5.0f;
            const float* pl = locs + ((size_t)b * P + p) * 4;
            accL += fabsf(pl[0] - g0) + fabsf(pl[1] - g1) +
                    fabsf(pl[2] - g2) + fabsf(pl[3] - g3);
            accC += bce;
            accN += 1.0f;
            s_row[p] = 0.0f;
        } else {
            s_row[p] = bce;
        }
    }
    __syncthreads();

    float confpos = blk_reduce(accC, s_redv);
    float l1sum   = blk_reduce(accL, s_redv);
    float npos    = blk_reduce(accN, s_redv);
    if (tid == 0) {
        ws[b]           = confpos;
        ws[128 + b]     = l1sum;
        ws[384 + b]     = npos;
    }

    // ---- Phase 4: exact top-k sum of conf_neg (k = min(3*npos, P)) ----
    // Non-negative floats are order-isomorphic to their bit patterns: binary
    // search the k-th largest value T, then sum = sum_{v>T} v + (k-cnt_{>T})*T.
    int k = 3 * (int)(npos + 0.5f);
    k = (k > P) ? P : k;
    unsigned cur = 0u;
    for (int bit = 30; bit >= 0; --bit) {
        unsigned cand = cur | (1u << bit);
        float t = __uint_as_float(cand);
        float lc = 0.0f;
        for (int p = tid; p < P; p += NT) lc += (s_row[p] >= t) ? 1.0f : 0.0f;
        float cnt = blk_reduce(lc, s_redv);
        if (cnt >= (float)k) cur = cand;   // uniform decision (broadcast from LDS)
    }
    float T = __uint_as_float(cur);
    float lg = 0.0f, ls = 0.0f;
    for (int p = tid; p < P; p += NT) {
        float v = s_row[p];
        if (v > T) { lg += 1.0f; ls += v; }
    }
    float cg = blk_reduce(lg, s_redv);
    float sg = blk_reduce(ls, s_redv);
    if (tid == 0) {
        float hard = (k > 0) ? (sg + ((float)k - cg) * T) : 0.0f;
        ws[256 + b] = hard;
    }
}

// Single wave32: cross-batch reductions via V_WMMA_F32_16X16X4_F32 with a
// ones B-matrix (D[m][n] = rowsum(A_m) + C[m][n]); two chained 64-elem tiles
// per quantity. EXEC is all ones around the WMMA ops.
__global__ void __launch_bounds__(32)
finalize_kernel(const float* __restrict__ ws, float* __restrict__ out) {
    const int lane = threadIdx.x;
    v2f ones; ones.x = 1.0f; ones.y = 1.0f;
    float sums[4];
#pragma unroll
    for (int q = 0; q < 4; ++q) {
        const float* base = ws + q * 128;
        v8f c = {0.f, 0.f, 0.f, 0.f, 0.f, 0.f, 0.f, 0.f};
#pragma unroll
        for (int t = 0; t < 2; ++t) {
            const float* v = base + t * 64;
            v2f a;
            // A-matrix 16x4 f32 layout: lanes 0-15 hold K=0,1; lanes 16-31 hold K=2,3
            if (lane < 16) { a.x = v[lane * 4 + 0];        a.y = v[lane * 4 + 1]; }
            else           { a.x = v[(lane - 16) * 4 + 2]; a.y = v[(lane - 16) * 4 + 3]; }
            c = __builtin_amdgcn_wmma_f32_16x16x4_f32(false, a, false, ones,
                                                      (short)0, c, false, false);
        }
        // Column N=(lane%16) of D holds all 16 row-sums split across lane and lane^16.
        float s = c[0] + c[1] + c[2] + c[3] + c[4] + c[5] + c[6] + c[7];
        s += __shfl_xor(s, 16, 32);
        sums[q] = s;
    }
    float conf_pos = sums[0], l1s = sums[1], hard = sums[2], npt = sums[3];
    float loc  = (npt > 0.0f) ? (l1s / (4.0f * fmaxf(npt, 1.0f))) : 0.0f;
    float conf = (hard + conf_pos) / (1e-10f + npt);
    if (lane == 0) {
        out[0] = conf + 1.0f * loc;  // total (ALPHA = 1.0)
        out[1] = conf;
        out[2] = loc;
    }
    for (int i = lane; i < BATCH; i += 32) out[3 + i] = ws[384 + i];
}

extern "C" void kernel_launch(void* const* d_in, const int* in_sizes, int n_in,
                              void* d_out, int out_size, void* d_ws, size_t ws_size,
                              hipStream_t stream) {
    (void)in_sizes; (void)n_in; (void)out_size; (void)ws_size;
    const float* locs   = (const float*)d_in[0];  // (B,P,4)
    const float* scores = (const float*)d_in[1];  // (B,P,3)
    const float* boxes  = (const float*)d_in[2];  // (B,M,4)
    const int*   labels = (const int*)d_in[3];    // (B,M)
    const float* priors = (const float*)d_in[4];  // (P,4)
    float* ws  = (float*)d_ws;
    float* out = (float*)d_out;

    match_loss_kernel<<<BATCH, NT, 0, stream>>>(locs, scores, boxes, labels, priors, ws);
    finalize_kernel<<<1, 32, 0, stream>>>(ws, out);
}